// RegionLossV2_62921270886753
// MI455X (gfx1250) — compile-verified
//
#include <hip/hip_runtime.h>
#include <hip/hip_bf16.h>
#include <math.h>

typedef __attribute__((ext_vector_type(2))) float v2f;
typedef __attribute__((ext_vector_type(8))) float v8f;

#define NBv 64
#define NAv 5
#define NCv 20
#define NHv 38
#define NWv 38
#define TMAXv 50
#define OUT_CH (5 + NCv)            // 25
#define CELLS_PER_B (NAv * NHv * NWv)  // 7220
#define CHAN_STRIDE (NHv * NWv)     // 1444

__device__ __forceinline__ float sigmoidf_(float v) {
    return 1.0f / (1.0f + __expf(-v));
}

__device__ __forceinline__ float iou_cxcywh(float ax, float ay, float aw, float ah,
                                            float bx, float by, float bw, float bh) {
    float mx = fminf(ax - aw * 0.5f, bx - bw * 0.5f);
    float Mx = fmaxf(ax + aw * 0.5f, bx + bw * 0.5f);
    float my = fminf(ay - ah * 0.5f, by - bh * 0.5f);
    float My = fmaxf(ay + ah * 0.5f, by + bh * 0.5f);
    float cw = aw + bw - (Mx - mx);
    float ch = ah + bh - (My - my);
    float inter = (cw > 0.0f && ch > 0.0f) ? cw * ch : 0.0f;
    float uni = aw * ah + bw * bh - inter;
    return inter / uni;
}

__global__ void region_zero_kernel(float* out) {
    if (threadIdx.x == 0 && blockIdx.x == 0) out[0] = 0.0f;
}

__global__ __launch_bounds__(256) void region_loss_kernel(
    const float* __restrict__ out, const float* __restrict__ target,
    const float* __restrict__ anchors, const int* __restrict__ seen_p,
    float* __restrict__ loss_out)
{
    __shared__ float s_gx[TMAXv], s_gy[TMAXv], s_gw[TMAXv], s_gh[TMAXv];
    __shared__ float s_txf[TMAXv], s_tyf[TMAXv], s_twv[TMAXv], s_thv[TMAXv], s_iou[TMAXv];
    __shared__ int   s_bn[TMAXv], s_gi[TMAXv], s_gj[TMAXv], s_val[TMAXv];
    __shared__ float sdata[256];
    __shared__ float red16[16];

    const int b   = blockIdx.y;
    const int tid = threadIdx.x;
    const int seen = seen_p[0];

    // ---- Stage per-target info for this batch image into LDS (threads 0..49) ----
    if (tid < TMAXv) {
        const float* tg = target + (b * TMAXv + tid) * 5;
        float gx = tg[1] * (float)NWv;
        float gy = tg[2] * (float)NHv;
        float gw = tg[3] * (float)NWv;
        float gh = tg[4] * (float)NHv;

        // validity = cumprod(target_x > 0) up to this index (prefix AND)
        int valid = 1;
        for (int t2 = 0; t2 <= tid; ++t2)
            valid &= (target[(b * TMAXv + t2) * 5 + 1] > 0.0f) ? 1 : 0;

        // anchor-shape argmax (first max wins, like jnp.argmax)
        int bn = 0; float best = -1.0f;
        for (int a = 0; a < NAv; ++a) {
            float aw = anchors[2 * a], ah = anchors[2 * a + 1];
            float iw = fminf(gw, aw), ih = fminf(gh, ah);
            float inter = iw * ih;
            float r = inter / (gw * gh + aw * ah - inter);
            if (r > best) { best = r; bn = a; }
        }
        int gi = (int)gx; gi = gi < 0 ? 0 : (gi > NWv - 1 ? NWv - 1 : gi);
        int gj = (int)gy; gj = gj < 0 ? 0 : (gj > NHv - 1 ? NHv - 1 : gj);

        float aw = anchors[2 * bn], ah = anchors[2 * bn + 1];
        float twv = __logf(fmaxf(gw, 1e-8f) / aw);
        float thv = __logf(fmaxf(gh, 1e-8f) / ah);

        // predicted box of responsible cell (b, bn, gj, gi)
        int base = (((b * NAv + bn) * OUT_CH) * NHv + gj) * NWv + gi;
        float px = sigmoidf_(out[base]) + (float)gi;
        float py = sigmoidf_(out[base + CHAN_STRIDE]) + (float)gj;
        float pw = __expf(out[base + 2 * CHAN_STRIDE]) * aw;
        float ph = __expf(out[base + 3 * CHAN_STRIDE]) * ah;

        s_gx[tid] = gx; s_gy[tid] = gy; s_gw[tid] = gw; s_gh[tid] = gh;
        s_txf[tid] = gx - (float)gi;
        s_tyf[tid] = gy - (float)gj;
        s_twv[tid] = twv; s_thv[tid] = thv;
        s_iou[tid] = iou_cxcywh(gx, gy, gw, gh, px, py, pw, ph);
        s_bn[tid] = bn; s_gi[tid] = gi; s_gj[tid] = gj; s_val[tid] = valid;
    }
    __syncthreads();

    // ---- Per-cell loss ----
    int cell = blockIdx.x * blockDim.x + tid;
    float loss = 0.0f;
    if (cell < CELLS_PER_B) {
        int i = cell % NWv;
        int j = (cell / NWv) % NHv;
        int a = cell / (NHv * NWv);

        int base = (((b * NAv + a) * OUT_CH) * NHv + j) * NWv + i;
        float x = sigmoidf_(out[base]);
        float y = sigmoidf_(out[base + CHAN_STRIDE]);
        float w = out[base + 2 * CHAN_STRIDE];
        float h = out[base + 3 * CHAN_STRIDE];
        float conf = sigmoidf_(out[base + 4 * CHAN_STRIDE]);

        float aw = anchors[2 * a], ah = anchors[2 * a + 1];
        float px = x + (float)i, py = y + (float)j;
        float pw = __expf(w) * aw, ph = __expf(h) * ah;

        // best IoU vs all valid ground-truth boxes (valid is a prefix -> break)
        float best_iou = 0.0f;
        for (int t = 0; t < TMAXv; ++t) {
            if (!s_val[t]) break;
            float u = iou_cxcywh(s_gx[t], s_gy[t], s_gw[t], s_gh[t], px, py, pw, ph);
            best_iou = fmaxf(best_iou, u);
        }

        float cm_conf = (best_iou > 0.6f) ? 0.0f : 1.0f;  // NOOBJECT_SCALE
        float cm_coord, tx, ty;
        if (seen < 12800) { cm_coord = 1.0f; tx = 0.5f; ty = 0.5f; }
        else              { cm_coord = 0.0f; tx = 0.0f; ty = 0.0f; }
        float tw = 0.0f, th = 0.0f, tconf = 0.0f;

        // scatter targets landing on this cell (last write wins, mirrors .at[].set)
        for (int t = 0; t < TMAXv; ++t) {
            if (!s_val[t]) break;
            if (s_bn[t] == a && s_gj[t] == j && s_gi[t] == i) {
                cm_coord = 1.0f;
                cm_conf  = 5.0f;  // OBJECT_SCALE
                tx = s_txf[t]; ty = s_tyf[t];
                tw = s_twv[t]; th = s_thv[t];
                tconf = s_iou[t];
            }
        }

        float dx = x - tx, dy = y - ty, dw = w - tw, dh = h - th, dc = conf - tconf;
        // masks are {0,1,5}: (mask*d)^2 == mask*d^2 for coord (0/1); (sqrt(m)*d)^2 == m*d^2 for conf
        loss = 0.5f * (cm_coord * (dx * dx + dy * dy + dw * dw + dh * dh)
                       + cm_conf * dc * dc);
    }

    // ---- Block reduction: 256 -> 32 via LDS tree ----
    sdata[tid] = loss;
    __syncthreads();
    for (int s = 128; s >= 32; s >>= 1) {
        if (tid < s) sdata[tid] += sdata[tid + s];
        __syncthreads();
    }

    // ---- 32 -> 16 via f32 WMMA: D = ones(16x4) x B(4x16), column sums are exact f32 sums ----
    if (tid < 32) {
        float p = sdata[tid];
        v2f av = {1.0f, 1.0f};       // A = all-ones 16x4 (2 VGPRs/lane)
        v2f bv = {p, 0.0f};          // B carries the 32 partials in slot 0
        v8f c = {};
        c = __builtin_amdgcn_wmma_f32_16x16x4_f32(
            /*neg_a=*/false, av, /*neg_b=*/false, bv,
            /*c_mod=*/(short)0, c, /*reuse_a=*/false, /*reuse_b=*/false);
        if (tid < 16) red16[tid] = c[0];   // D[0][n] = column sum n
    }
    __syncthreads();
    if (tid == 0) {
        float s = 0.0f;
        for (int k = 0; k < 16; ++k) s += red16[k];
        atomicAdd(loss_out, s);
    }
}

extern "C" void kernel_launch(void* const* d_in, const int* in_sizes, int n_in,
                              void* d_out, int out_size, void* d_ws, size_t ws_size,
                              hipStream_t stream) {
    (void)in_sizes; (void)n_in; (void)d_ws; (void)ws_size; (void)out_size;
    const float* output  = (const float*)d_in[0];
    const float* target  = (const float*)d_in[1];
    const float* anchors = (const float*)d_in[2];
    const int*   seen    = (const int*)d_in[3];
    float* loss = (float*)d_out;

    region_zero_kernel<<<1, 1, 0, stream>>>(loss);

    dim3 grid((CELLS_PER_B + 255) / 256, NBv);
    region_loss_kernel<<<grid, 256, 0, stream>>>(output, target, anchors, seen, loss);
}